// RegionAttNew_42623255446294
// MI455X (gfx1250) — compile-verified
//
#include <hip/hip_runtime.h>

#define B_    4
#define C_    512
#define H_    64
#define W_    208
#define HID_  512
#define DS_   16
#define NREG_ 4
#define W4_   (W_ / 4)        // 52 vec4 groups per row

typedef __attribute__((ext_vector_type(2))) float v2f;
typedef __attribute__((ext_vector_type(4))) float v4f;
typedef __attribute__((ext_vector_type(8))) float v8f;
typedef __attribute__((ext_vector_type(4))) int   v4i;

// ---------------------------------------------------------------------------
// Kernel A: per-batch stable counting sort of column labels.
// col_labels[b][w] = text_mask[b,0,0,w*DS]; gather[b][pos] = src_w | (r<<16)
// ---------------------------------------------------------------------------
__global__ void RegionAtt_order_kernel(const int* __restrict__ text_mask,
                                       int* __restrict__ gather)
{
    int b = threadIdx.x;
    if (b >= B_) return;
    const int* base = text_mask + (size_t)b * (H_ * DS_) * (W_ * DS_);
    int cnt[NREG_] = {0, 0, 0, 0};
    for (int w = 0; w < W_; ++w) {
        int l = base[w * DS_] - 1;          // row 0, col w*DS
        cnt[l]++;
    }
    int off[NREG_];
    off[0] = 0;
    for (int i = 1; i < NREG_; ++i) off[i] = off[i - 1] + cnt[i - 1];
    for (int w = 0; w < W_; ++w) {          // in-order scan => stable
        int l = base[w * DS_] - 1;
        int pos = off[l]++;
        gather[b * W_ + pos] = w | (l << 16);
    }
}

// ---------------------------------------------------------------------------
// Kernel B: delta[b][r][c] = (text_feat[r] @ Wv[r] @ Wo[r])[b][c]
// One workgroup per r, 32 waves; wave nw owns 16-column tile n0 = nw*16.
// A (M=4) is zero-padded to M=16 in LDS so the WMMA inner loops are
// branch-free.  f32 WMMA 16x16x4 matches reference f32 precision (RNE).
// ---------------------------------------------------------------------------
__global__ __launch_bounds__(1024) void RegionAtt_delta_wmma_kernel(
    const float* __restrict__ text_feat,   // (NREG, B, HID)
    const float* __restrict__ Wv,          // (NREG, HID, HID)
    const float* __restrict__ Wo,          // (NREG, HID, C)
    float* __restrict__ delta)             // (B, NREG, C)
{
    __shared__ float tfs[16][HID_ + 4];    // zero-padded A for stage 1
    __shared__ float tmp[16][HID_ + 4];    // stage-1 result (padded vs bank conflicts)

    const int r    = blockIdx.x;
    const int tid  = threadIdx.x;
    const int lane = tid & 31;
    const int nw   = tid >> 5;             // wave id -> 16-column tile
    const int lo   = lane & 15;            // M for A, N for B/C/D
    const int hi   = lane >> 4;            // K-half selector (VGPR layout)
    const int n0   = nw * 16;

    const float* tf  = text_feat + r * (B_ * HID_);
    const float* wvp = Wv + r * (HID_ * HID_);
    const float* wop = Wo + r * (HID_ * C_);

    // Stage A into LDS with zero padding: rows 0..3 = text_feat[r], 4..15 = 0
    for (int i = tid; i < 16 * HID_; i += 1024) {
        int m = i >> 9;                    // /HID_
        int k = i & (HID_ - 1);
        tfs[m][k] = (m < B_) ? tf[m * HID_ + k] : 0.0f;
    }
    __syncthreads();

    // ---- stage 1: tmp(16 x 512) = Apad(16 x 512) @ Wv[r](512 x 512) tile ----
    v8f acc = {};
    for (int k0 = 0; k0 < HID_; k0 += 4) {
        const int ka = k0 + hi * 2;
        v2f a, b;
        a.x = tfs[lo][ka];
        a.y = tfs[lo][ka + 1];
        b.x = wvp[(ka    ) * HID_ + n0 + lo];
        b.y = wvp[(ka + 1) * HID_ + n0 + lo];
        acc = __builtin_amdgcn_wmma_f32_16x16x4_f32(false, a, false, b,
                                                    (short)0, acc, false, false);
    }
#pragma unroll
    for (int v = 0; v < 8; ++v)
        tmp[v + 8 * hi][n0 + lo] = acc[v];
    __syncthreads();

    // ---- stage 2: delta tile = tmp(16 x 512) @ Wo[r](512 x 512) tile ----
    v8f acc2 = {};
    for (int k0 = 0; k0 < HID_; k0 += 4) {
        const int ka = k0 + hi * 2;
        v2f a, b;
        a.x = tmp[lo][ka];
        a.y = tmp[lo][ka + 1];
        b.x = wop[(ka    ) * C_ + n0 + lo];
        b.y = wop[(ka + 1) * C_ + n0 + lo];
        acc2 = __builtin_amdgcn_wmma_f32_16x16x4_f32(false, a, false, b,
                                                     (short)0, acc2, false, false);
    }
#pragma unroll
    for (int v = 0; v < 8; ++v) {
        int m = v + 8 * hi;
        if (m < B_)                         // only the 4 real batch rows
            delta[(m * NREG_ + r) * C_ + n0 + lo] = acc2[v];
    }
}

// ---------------------------------------------------------------------------
// Kernel C: out[b,c,h,4w..4w+3] = img[b,c,h,src_i] + delta[b,r_i,c]
// Vectorized 4-wide along w: one b128 gather-table load, four gathered img
// loads, one b128 nontemporal store.  218 MB streaming @ 23.3 TB/s ~ 9.4 us.
// ---------------------------------------------------------------------------
__global__ __launch_bounds__(256) void RegionAtt_apply_kernel(
    const float* __restrict__ img,
    const float* __restrict__ delta,
    const int*  __restrict__ gather,
    float* __restrict__ out, int total4)
{
    int idx = blockIdx.x * 256 + threadIdx.x;      // vec4 index
    if (idx >= total4) return;
    int w4 = idx % W4_;
    int t  = idx / W4_;
    int h = t % H_;  t /= H_;
    int c = t % C_;
    int b = t / C_;

    v4i g = *(const v4i*)(gather + b * W_ + 4 * w4);   // 16B aligned, L2-hot

    const float* imrow = img   + ((size_t)(b * C_ + c) * H_ + h) * W_;
    const float* dcol  = delta + b * (NREG_ * C_) + c;  // + r*C_ selects region

    v4f o;
    o.x = __builtin_nontemporal_load(&imrow[g.x & 0xFFFF]) + dcol[(g.x >> 16) * C_];
    o.y = __builtin_nontemporal_load(&imrow[g.y & 0xFFFF]) + dcol[(g.y >> 16) * C_];
    o.z = __builtin_nontemporal_load(&imrow[g.z & 0xFFFF]) + dcol[(g.z >> 16) * C_];
    o.w = __builtin_nontemporal_load(&imrow[g.w & 0xFFFF]) + dcol[(g.w >> 16) * C_];

    __builtin_nontemporal_store(o, (v4f*)(out + (size_t)idx * 4));
}

// ---------------------------------------------------------------------------
extern "C" void kernel_launch(void* const* d_in, const int* in_sizes, int n_in,
                              void* d_out, int out_size, void* d_ws, size_t ws_size,
                              hipStream_t stream) {
    const float* img       = (const float*)d_in[0];  // image_feature (B,C,H,W)
    const float* text_feat = (const float*)d_in[1];  // (NREG,B,HID)
    const int*   text_mask = (const int*)  d_in[2];  // (B,1,H*DS,W*DS)
    // d_in[3] = Wq, d_in[4] = Wk : unused by the reference
    const float* Wv        = (const float*)d_in[5];  // (NREG,HID,HID)
    const float* Wo        = (const float*)d_in[6];  // (NREG,HID,C)
    float* out = (float*)d_out;

    float* delta  = (float*)d_ws;                      // 32 KB
    int*   gather = (int*)((char*)d_ws + 32768);       // 3.3 KB, 16B aligned

    RegionAtt_order_kernel<<<1, 32, 0, stream>>>(text_mask, gather);
    RegionAtt_delta_wmma_kernel<<<NREG_, 1024, 0, stream>>>(text_feat, Wv, Wo, delta);

    const int total4 = (B_ * C_ * H_ * W_) / 4;        // 6,815,744 vec4 groups
    RegionAtt_apply_kernel<<<(total4 + 255) / 256, 256, 0, stream>>>(
        img, delta, gather, out, total4);
}